// Model_64209761075485
// MI455X (gfx1250) — compile-verified
//
#include <hip/hip_runtime.h>
#include <hip/hip_bf16.h>
#include <math.h>

#define B_ 32
#define L_ 48
#define LA_ 8
#define D_ 256
#define H_ 256
#define PG_ 5
#define DH_ (D_ * H_)   // 65536

typedef float v2f __attribute__((ext_vector_type(2)));
typedef float v8f __attribute__((ext_vector_type(8)));

// ---------------------------------------------------------------- utilities
__device__ __forceinline__ float block_reduce_sum(float v, float* red) {
    // wave32 reduce
    #pragma unroll
    for (int o = 16; o > 0; o >>= 1) v += __shfl_down(v, o, 32);
    if ((threadIdx.x & 31) == 0) red[threadIdx.x >> 5] = v;
    __syncthreads();
    float s = 0.f;
    #pragma unroll
    for (int i = 0; i < 8; i++) s += red[i];
    __syncthreads();            // protect re-use of red
    return s;                   // full sum in every thread
}

// ---------------------------------------------------------------- embedding gather
__global__ void k_gather(const int* __restrict__ ids, const float* __restrict__ emb,
                         float* __restrict__ out, int ntok) {
    int idx = blockIdx.x * blockDim.x + threadIdx.x;
    if (idx >= ntok * D_) return;
    int tok = idx >> 8, d = idx & 255;
    out[idx] = emb[(size_t)ids[tok] * D_ + d];
}

// ---------------------------------------------------------------- WMMA f32 GEMM
// C[M,N] = A[M,K] @ B[K,N], all row-major f32. One wave per 16x16 tile,
// 8 waves (256 threads) per block. M,N,K multiples of 16 (K mult of 4).
__global__ void k_gemm_wmma(const float* __restrict__ A, const float* __restrict__ Bm,
                            float* __restrict__ C, int M, int N, int K) {
    const int wave = threadIdx.x >> 5;
    const int lane = threadIdx.x & 31;
    const int tilesN = N >> 4;
    const int numTiles = (M >> 4) * tilesN;
    const int tile = blockIdx.x * (blockDim.x >> 5) + wave;
    if (tile >= numTiles) return;           // wave-uniform: EXEC stays all ones
    const int tm = tile / tilesN, tn = tile % tilesN;
    const int r16  = lane & 15;
    const int koff = (lane >> 4) << 1;      // 0 for lanes 0-15, 2 for lanes 16-31
    const int row = tm * 16 + r16;          // A row (M)
    const int col = tn * 16 + r16;          // B/C col (N)

    v8f acc = {};
    const float* arow = A + (size_t)row * K;
    for (int kk = 0; kk < K; kk += 4) {
        v2f a, b;
        a[0] = arow[kk + koff];
        a[1] = arow[kk + koff + 1];
        b[0] = Bm[(size_t)(kk + koff) * N + col];
        b[1] = Bm[(size_t)(kk + koff + 1) * N + col];
        acc = __builtin_amdgcn_wmma_f32_16x16x4_f32(false, a, false, b,
                                                    (short)0, acc, false, false);
    }
    #pragma unroll
    for (int v = 0; v < 8; v++) {
        int m = tm * 16 + v + ((lane >> 4) ? 8 : 0);
        C[(size_t)m * N + col] = acc[v];
    }
}

// ---------------------------------------------------------------- attention (per b,l)
__global__ void k_attn(const float* __restrict__ cWc, const float* __restrict__ aWa,
                       const float* __restrict__ asp, const float* __restrict__ Vatt,
                       const float* __restrict__ cWm, const float* __restrict__ Wm_b,
                       const float* __restrict__ Vm, const float* __restrict__ Vm_b,
                       float* __restrict__ ctx_attn, float* __restrict__ m_logit) {
    __shared__ float red[8];
    const int bl = blockIdx.x;              // b*L + l
    const int b = bl / L_;
    const int tid = threadIdx.x;            // d index

    const float vc   = cWc[bl * D_ + tid];
    const float vatt = Vatt[tid];
    float sc[LA_];
    #pragma unroll
    for (int la = 0; la < LA_; la++) {
        float t = tanhf(vc + aWa[(b * LA_ + la) * D_ + tid]) * vatt;
        sc[la] = block_reduce_sum(t, red);  // full value in every thread
    }
    // softmax over aspect dim (redundant per thread)
    float mx = sc[0];
    #pragma unroll
    for (int la = 1; la < LA_; la++) mx = fmaxf(mx, sc[la]);
    float den = 0.f;
    #pragma unroll
    for (int la = 0; la < LA_; la++) { sc[la] = __expf(sc[la] - mx); den += sc[la]; }
    float inv = 1.f / den;
    float cv = 0.f;
    #pragma unroll
    for (int la = 0; la < LA_; la++)
        cv = fmaf(sc[la] * inv, asp[(b * LA_ + la) * D_ + tid], cv);
    ctx_attn[bl * D_ + tid] = cv;

    float m = tanhf(cWm[bl * D_ + tid] + Wm_b[tid] + cv) * Vm[tid];
    float ms = block_reduce_sum(m, red);
    if (tid == 0) m_logit[bl] = ms + Vm_b[0];
}

// ---------------------------------------------------------------- softmax over L (per b)
__global__ void k_softmax_l(const float* __restrict__ m_logit, float* __restrict__ a2) {
    int b = blockIdx.x;
    if (threadIdx.x != 0) return;
    float mx = -1e30f;
    for (int l = 0; l < L_; l++) mx = fmaxf(mx, m_logit[b * L_ + l]);
    float den = 0.f;
    float e[L_];
    for (int l = 0; l < L_; l++) { e[l] = __expf(m_logit[b * L_ + l] - mx); den += e[l]; }
    float inv = 1.f / den;
    for (int l = 0; l < L_; l++) a2[b * L_ + l] = e[l] * inv;
}

// ---------------------------------------------------------------- param generator (per dir,b,l)
// hid_* = relu(x @ w1 + b1) (5-vec);  Bt = relu(hid_b @ wb2 + bb2);
// xWx[h] = sum_d x[d] * relu(sum_p hid_x[p]*wx2[p, d*H+h] + bx2[d*H+h])
__global__ void k_pg(const float* __restrict__ ctx,
                     const float* __restrict__ wx_w1, const float* __restrict__ wx_b1,
                     const float* __restrict__ wx_w2, const float* __restrict__ wx_b2,
                     const float* __restrict__ wh_w1, const float* __restrict__ wh_b1,
                     const float* __restrict__ wb_w1, const float* __restrict__ wb_b1,
                     const float* __restrict__ wb_w2, const float* __restrict__ wb_b2,
                     float* __restrict__ hid_wh, float* __restrict__ xWx,
                     float* __restrict__ Bt) {
    __shared__ float xs[D_];
    __shared__ float red[8];
    const int idx = blockIdx.x;                 // dir*(B*L) + bl
    const int dir = idx / (B_ * L_);
    const int bl  = idx % (B_ * L_);
    const int tid = threadIdx.x;

    const float x = ctx[bl * D_ + tid];
    xs[tid] = x;
    __syncthreads();

    const float* w1x = wx_w1 + dir * D_ * PG_;
    const float* w1h = wh_w1 + dir * D_ * PG_;
    const float* w1b = wb_w1 + dir * D_ * PG_;
    float hx[PG_], hh[PG_], hb[PG_];
    #pragma unroll
    for (int p = 0; p < PG_; p++) {
        hx[p] = fmaxf(block_reduce_sum(x * w1x[tid * PG_ + p], red) + wx_b1[dir * PG_ + p], 0.f);
        hh[p] = fmaxf(block_reduce_sum(x * w1h[tid * PG_ + p], red) + wh_b1[dir * PG_ + p], 0.f);
        hb[p] = fmaxf(block_reduce_sum(x * w1b[tid * PG_ + p], red) + wb_b1[dir * PG_ + p], 0.f);
    }
    if (tid < PG_) hid_wh[idx * PG_ + tid] = hh[tid];

    // Bt (h index = tid)
    const float* w2b = wb_w2 + dir * PG_ * H_;
    float bt = wb_b2[dir * H_ + tid];
    #pragma unroll
    for (int p = 0; p < PG_; p++) bt = fmaf(hb[p], w2b[p * H_ + tid], bt);
    Bt[idx * H_ + tid] = fmaxf(bt, 0.f);

    // fused Wx generation + contraction with x  (h index = tid, loop over d)
    const float* w2x = wx_w2 + (size_t)dir * PG_ * DH_;
    const float* b2x = wx_b2 + (size_t)dir * DH_;
    float acc = 0.f;
    for (int d = 0; d < D_; d++) {
        int o = d * H_ + tid;
        float w = b2x[o];
        w = fmaf(hx[0], w2x[o], w);
        w = fmaf(hx[1], w2x[DH_ + o], w);
        w = fmaf(hx[2], w2x[2 * DH_ + o], w);
        w = fmaf(hx[3], w2x[3 * DH_ + o], w);
        w = fmaf(hx[4], w2x[4 * DH_ + o], w);
        acc = fmaf(xs[d], fmaxf(w, 0.f), acc);
    }
    xWx[idx * H_ + tid] = acc;
}

// ---------------------------------------------------------------- recurrence (per dir,b)
// h_t = relu(xWx_t + h_{t-1} @ Wh_t + Bt_t), Wh_t generated on the fly.
__global__ void k_rnn(const float* __restrict__ hid_wh, const float* __restrict__ xWx,
                      const float* __restrict__ Bt,
                      const float* __restrict__ wh_w2, const float* __restrict__ wh_b2,
                      float* __restrict__ senti) {
    __shared__ float hs[H_];
    __shared__ float hid[PG_];
    const int dir = blockIdx.x >> 5;   // 32 blocks per direction
    const int b   = blockIdx.x & 31;
    const int tid = threadIdx.x;       // output index k

    hs[tid] = 0.f;
    const float* w2 = wh_w2 + (size_t)dir * PG_ * DH_;
    const float* b2 = wh_b2 + (size_t)dir * DH_;

    for (int s = 0; s < L_; s++) {
        const int t = dir ? (L_ - 1 - s) : s;
        const int idx = dir * (B_ * L_) + b * L_ + t;
        if (tid < PG_) hid[tid] = hid_wh[idx * PG_ + tid];
        __syncthreads();

        float acc = xWx[idx * H_ + tid] + Bt[idx * H_ + tid];
        if (s > 0) {
            const float h0 = hid[0], h1 = hid[1], h2 = hid[2], h3 = hid[3], h4 = hid[4];
            for (int i = 0; i < H_; i++) {       // i = h_in
                int o = i * H_ + tid;
                float w = b2[o];
                w = fmaf(h0, w2[o], w);
                w = fmaf(h1, w2[DH_ + o], w);
                w = fmaf(h2, w2[2 * DH_ + o], w);
                w = fmaf(h3, w2[3 * DH_ + o], w);
                w = fmaf(h4, w2[4 * DH_ + o], w);
                acc = fmaf(hs[i], fmaxf(w, 0.f), acc);
            }
        }
        float hn = fmaxf(acc, 0.f);
        __syncthreads();
        hs[tid] = hn;
        senti[(size_t)(b * L_ + t) * (2 * H_) + dir * H_ + tid] = hn;
    }
}

// ---------------------------------------------------------------- pooled @ Wout + bout (per b)
__global__ void k_pool_out(const float* __restrict__ a2, const float* __restrict__ senti,
                           const float* __restrict__ Wout, const float* __restrict__ bout,
                           float* __restrict__ out) {
    __shared__ float red[8];
    const int b = blockIdx.x;
    const int tid = threadIdx.x;
    float p0 = 0.f, p1 = 0.f;
    for (int l = 0; l < L_; l++) {
        float a = a2[b * L_ + l];
        const float* srow = senti + (size_t)(b * L_ + l) * (2 * H_);
        p0 = fmaf(a, srow[tid], p0);
        p1 = fmaf(a, srow[H_ + tid], p1);
    }
    #pragma unroll
    for (int c = 0; c < 3; c++) {
        float part = p0 * Wout[tid * 3 + c] + p1 * Wout[(H_ + tid) * 3 + c];
        float s = block_reduce_sum(part, red);
        if (tid == 0) out[b * 3 + c] = s + bout[c];
    }
}

// ---------------------------------------------------------------- launch
extern "C" void kernel_launch(void* const* d_in, const int* in_sizes, int n_in,
                              void* d_out, int out_size, void* d_ws, size_t ws_size,
                              hipStream_t stream) {
    const int*   context_ids = (const int*)  d_in[0];
    const int*   aspect_ids  = (const int*)  d_in[2];
    const float* emb   = (const float*)d_in[4];
    const float* Wc    = (const float*)d_in[5];
    const float* Wa    = (const float*)d_in[6];
    const float* Vatt  = (const float*)d_in[7];
    const float* Wm    = (const float*)d_in[8];
    const float* Wm_b  = (const float*)d_in[9];
    const float* Vm    = (const float*)d_in[10];
    const float* Vm_b  = (const float*)d_in[11];
    const float* Wout  = (const float*)d_in[12];
    const float* bout  = (const float*)d_in[13];
    const float* wx_w1 = (const float*)d_in[14];
    const float* wx_b1 = (const float*)d_in[15];
    const float* wx_w2 = (const float*)d_in[16];
    const float* wx_b2 = (const float*)d_in[17];
    const float* wh_w1 = (const float*)d_in[18];
    const float* wh_b1 = (const float*)d_in[19];
    const float* wh_w2 = (const float*)d_in[20];
    const float* wh_b2 = (const float*)d_in[21];
    const float* wb_w1 = (const float*)d_in[22];
    const float* wb_b1 = (const float*)d_in[23];
    const float* wb_w2 = (const float*)d_in[24];
    const float* wb_b2 = (const float*)d_in[25];
    float* out = (float*)d_out;

    float* ws = (float*)d_ws;
    const size_t N_CTX = (size_t)B_ * L_ * D_;       // 393216
    const size_t N_ASP = (size_t)B_ * LA_ * D_;      // 65536
    float* ctx      = ws;               ws += N_CTX;
    float* asp      = ws;               ws += N_ASP;
    float* cWc      = ws;               ws += N_CTX;
    float* aWa      = ws;               ws += N_ASP;
    float* cWm      = ws;               ws += N_CTX;
    float* ctx_attn = ws;               ws += N_CTX;
    float* m_logit  = ws;               ws += B_ * L_;
    float* a2       = ws;               ws += B_ * L_;
    float* hid_wh   = ws;               ws += 2 * B_ * L_ * PG_;
    float* xWx      = ws;               ws += 2 * N_CTX;
    float* Btb      = ws;               ws += 2 * N_CTX;
    float* senti    = ws;               ws += 2 * N_CTX;

    // 1) embedding gathers
    k_gather<<<(B_ * L_ * D_) / 256, 256, 0, stream>>>(context_ids, emb, ctx, B_ * L_);
    k_gather<<<(B_ * LA_ * D_) / 256, 256, 0, stream>>>(aspect_ids, emb, asp, B_ * LA_);

    // 2) WMMA GEMMs: ctx@Wc, asp@Wa, ctx@Wm
    {
        int tilesCtx = (B_ * L_ / 16) * (D_ / 16);   // 1536 -> 192 blocks of 8 waves
        int tilesAsp = (B_ * LA_ / 16) * (D_ / 16);  // 256  -> 32 blocks
        k_gemm_wmma<<<(tilesCtx + 7) / 8, 256, 0, stream>>>(ctx, Wc, cWc, B_ * L_, D_, D_);
        k_gemm_wmma<<<(tilesAsp + 7) / 8, 256, 0, stream>>>(asp, Wa, aWa, B_ * LA_, D_, D_);
        k_gemm_wmma<<<(tilesCtx + 7) / 8, 256, 0, stream>>>(ctx, Wm, cWm, B_ * L_, D_, D_);
    }

    // 3) attention + token-weight logits
    k_attn<<<B_ * L_, 256, 0, stream>>>(cWc, aWa, asp, Vatt, cWm, Wm_b, Vm, Vm_b,
                                        ctx_attn, m_logit);
    // 4) softmax over sequence dim
    k_softmax_l<<<B_, 32, 0, stream>>>(m_logit, a2);

    // 5) param generator: hidden vecs, Bt, fused xWx for both directions
    k_pg<<<2 * B_ * L_, 256, 0, stream>>>(ctx,
                                          wx_w1, wx_b1, wx_w2, wx_b2,
                                          wh_w1, wh_b1,
                                          wb_w1, wb_b1, wb_w2, wb_b2,
                                          hid_wh, xWx, Btb);

    // 6) bidirectional recurrence with on-the-fly Wh generation
    k_rnn<<<2 * B_, 256, 0, stream>>>(hid_wh, xWx, Btb, wh_w2, wh_b2, senti);

    // 7) attention pooling + classifier head
    k_pool_out<<<B_, 256, 0, stream>>>(a2, senti, Wout, bout, out);
}